// HungarianMatcher_37306085933291
// MI455X (gfx1250) — compile-verified
//
#include <hip/hip_runtime.h>
#include <stdint.h>

#define BS   16
#define NQ   64
#define NGT  32
#define DIM  4096
#define NCLS 33
#define DC   128   // D-chunk staged in LDS per iteration
#define RS   130   // LDS row stride (floats): even -> 8B-aligned b64 reads, bank-spread residue

// ---- CDNA5 async global->LDS copy (ASYNCcnt-tracked), per-lane addresses ----
__device__ __forceinline__ void async_ld_b32(uint32_t lds_byte_addr, const float* gsrc) {
    asm volatile("global_load_async_to_lds_b32 %0, %1, off"
                 :: "v"(lds_byte_addr), "v"(gsrc) : "memory");
}
__device__ __forceinline__ void wait_async0() {
    asm volatile("s_wait_asynccnt 0x0" ::: "memory");
}

// ============================================================================
// Kernel 1: partial L1 cost blocks over a D-slice.
//   part[s][b][g][q] = sum_{d in slice s} |heat[b,q,d] - gt[b,g,d]|
// Grid (BS, nslices); one workgroup (256 thr = 8 waves) per (batch, slice).
// 2x4 (q,g) register tile per thread; LDS staged via async-to-LDS.
// ============================================================================
__global__ __launch_bounds__(256) void cost_kernel(
    const float* __restrict__ heat, const float* __restrict__ gt,
    float* __restrict__ part, int slice_len)
{
    __shared__ float hs[NQ  * RS];   // 64*130*4  = 33280 B, layout [q][d]
    __shared__ float gs[NGT * RS];   // 32*130*4  = 16640 B, layout [g][d]

    const int b    = blockIdx.x;
    const int s    = blockIdx.y;
    const int tid  = threadIdx.x;
    const int lane = tid & 31;       // wave32
    const int w    = tid >> 5;       // 8 waves
    const int q0 = lane, q1 = lane + 32;   // lane's two query rows
    const int g0 = w * 4;                  // wave's four gt rows (uniform per wave)

    const float* heatB = heat + (size_t)b * NQ  * DIM;
    const float* gtB   = gt   + (size_t)b * NGT * DIM;

    float acc[2][4];
#pragma unroll
    for (int a = 0; a < 2; ++a)
#pragma unroll
        for (int c = 0; c < 4; ++c) acc[a][c] = 0.f;

    const int dfill = tid & (DC - 1);   // lanes -> consecutive d: coalesced global, conflict-free LDS
    const int rbase = tid >> 7;         // 0 or 1

    const int dbeg = s * slice_len;
    const int dend = dbeg + slice_len;

    for (int d0 = dbeg; d0 < dend; d0 += DC) {
        __syncthreads();  // prior chunk fully consumed before overwrite

        // stage heat chunk: 64 rows x 128 d  (32 async b32 per thread)
#pragma unroll
        for (int k = 0; k < NQ / 2; ++k) {
            int q = rbase + 2 * k;
            async_ld_b32((uint32_t)(uintptr_t)&hs[q * RS + dfill],
                         heatB + (size_t)q * DIM + d0 + dfill);
        }
        // stage gt chunk: 32 rows x 128 d  (16 async b32 per thread)
#pragma unroll
        for (int k = 0; k < NGT / 2; ++k) {
            int g = rbase + 2 * k;
            async_ld_b32((uint32_t)(uintptr_t)&gs[g * RS + dfill],
                         gtB + (size_t)g * DIM + d0 + dfill);
        }
        wait_async0();
        __syncthreads();

        // compute: per 2 d's -> 6 ds_load_b64 feeding 16 abs-diff accumulates
#pragma unroll 4
        for (int d = 0; d < DC; d += 2) {
            float2 h0 = *(const float2*)&hs[q0 * RS + d];
            float2 h1 = *(const float2*)&hs[q1 * RS + d];
#pragma unroll
            for (int c = 0; c < 4; ++c) {
                float2 gv = *(const float2*)&gs[(g0 + c) * RS + d];  // wave-uniform broadcast
                acc[0][c] += fabsf(h0.x - gv.x) + fabsf(h0.y - gv.y);
                acc[1][c] += fabsf(h1.x - gv.x) + fabsf(h1.y - gv.y);
            }
        }
    }

    // store partial block part[s][b][g][q] (no prob term here; added once in stage 2)
    float* dst = part + ((size_t)s * BS + b) * NGT * NQ;
#pragma unroll
    for (int c = 0; c < 4; ++c) {
        dst[(size_t)(g0 + c) * NQ + q0] = acc[0][c];
        dst[(size_t)(g0 + c) * NQ + q1] = acc[1][c];
    }
}

// ============================================================================
// Kernel 2: exact rectangular LSAP (Jonker-Volgenant), one wave32 per batch.
// Deterministically sums slice partials + class-prob term during load, then
// solves n = 32 rows x m = 64 cols; each lane owns columns {lane, lane+32}.
// ============================================================================
__global__ __launch_bounds__(32) void lsap_kernel(
    const float* __restrict__ part, int nslices,
    const float* __restrict__ prob, const int* __restrict__ label,
    int* __restrict__ out)
{
    const int b    = blockIdx.x;
    const int lane = threadIdx.x;

    __shared__ float cost[NGT * NQ];   // 8 KB
    __shared__ float u[NGT];
    __shared__ float shortest_s[NQ];
    __shared__ int   path_s[NQ];
    __shared__ int   row4col[NQ];
    __shared__ int   col4row[NGT];

    const float* probB  = prob  + (size_t)b * NQ * NCLS;
    const int*   labelB = label + b * NGT;

    // cost[g*64+q] = sum_s part[s][b][g][q] + 1 - prob[b,q,label[b,g]]
    for (int t = lane; t < NGT * NQ; t += 32) {
        float c = 0.f;
        for (int s = 0; s < nslices; ++s)          // fixed order -> deterministic
            c += part[((size_t)s * BS + b) * NGT * NQ + t];
        int g = t >> 6, q = t & 63;
        cost[t] = c + 1.0f - probB[q * NCLS + labelB[g]];
    }
    u[lane] = 0.f;
    col4row[lane] = -1;
    row4col[lane] = -1;
    row4col[lane + 32] = -1;
    float v0 = 0.f, v1 = 0.f;          // duals of lane's two columns
    __syncthreads();

    const float INF = __builtin_inff();

    for (int cur = 0; cur < NGT; ++cur) {
        float sh0 = INF, sh1 = INF;    // shortest[] for lane's columns
        int   p0 = 0, p1 = 0;          // path[] for lane's columns
        unsigned long long SC = 0ull;  // column-scanned mask (uniform)
        unsigned SR = 0u;              // row-scanned mask (uniform)
        float minVal = 0.f;
        int i = cur, sink = -1;

        for (int iter = 0; iter < NQ && sink < 0; ++iter) {
            SR |= 1u << i;
            float ui = u[i];
            if (!((SC >> lane) & 1ull)) {
                float low = minVal + cost[i * NQ + lane] - ui - v0;
                if (low < sh0) { sh0 = low; p0 = i; }
            }
            if (!((SC >> (lane + 32)) & 1ull)) {
                float low = minVal + cost[i * NQ + lane + 32] - ui - v1;
                if (low < sh1) { sh1 = low; p1 = i; }
            }
            // masked argmin over 64 columns, first-index tie-break (matches argmin)
            float c0 = ((SC >> lane) & 1ull) ? INF : sh0;
            float c1 = ((SC >> (lane + 32)) & 1ull) ? INF : sh1;
            float bv; int bj;
            if (c1 < c0) { bv = c1; bj = lane + 32; } else { bv = c0; bj = lane; }
#pragma unroll
            for (int off = 16; off > 0; off >>= 1) {
                float ov = __shfl_xor(bv, off, 32);
                int   oj = __shfl_xor(bj, off, 32);
                if (ov < bv || (ov == bv && oj < bj)) { bv = ov; bj = oj; }
            }
            minVal = bv;
            SC |= 1ull << bj;
            int r = row4col[bj];       // wave-uniform LDS read
            if (r < 0) sink = bj; else i = r;
        }

        // publish column-local state for cross-lane gathers
        shortest_s[lane]      = sh0;  shortest_s[lane + 32] = sh1;
        path_s[lane]          = p0;   path_s[lane + 32]     = p1;
        __syncthreads();

        // dual updates: lane == row for u; lane's own columns for v
        if ((SR >> lane) & 1u) {
            if (lane == cur) u[lane] += minVal;
            else             u[lane] += minVal - shortest_s[col4row[lane]];
        }
        if ((SC >> lane) & 1ull)        v0 -= minVal - sh0;
        if ((SC >> (lane + 32)) & 1ull) v1 -= minVal - sh1;
        __syncthreads();

        // augment alternating path back from sink (serial, lane 0)
        if (lane == 0) {
            int j = sink;
            while (true) {
                int ii = path_s[j];
                row4col[j] = ii;
                int prev = col4row[ii];
                col4row[ii] = j;
                if (ii == cur) break;
                j = prev;
            }
        }
        __syncthreads();
    }

    // output: rows = matched query indices sorted ascending, cols = gt indices
    int key  = col4row[lane];          // distinct values -> rank = #smaller
    int rank = 0;
    for (int t = 0; t < NGT; ++t) rank += (col4row[t] < key) ? 1 : 0;
    out[b * NGT + rank]            = key;    // rows block  [bs, ngt]
    out[BS * NGT + b * NGT + rank] = lane;   // cols block  [bs, ngt]
}

// ============================================================================
extern "C" void kernel_launch(void* const* d_in, const int* in_sizes, int n_in,
                              void* d_out, int out_size, void* d_ws, size_t ws_size,
                              hipStream_t stream)
{
    const float* prob  = (const float*)d_in[0];   // [16,64,33]
    const int*   label = (const int*)  d_in[1];   // [16,32]
    const float* heat  = (const float*)d_in[2];   // [16,64,4096]
    const float* gt    = (const float*)d_in[3];   // [16,32,4096]
    float* part = (float*)d_ws;                   // [nslices,16,32,64] partial cost blocks
    int*   out  = (int*)d_out;                    // 2 * 16 * 32 int32

    // largest power-of-two slice count (<=16) whose partial buffers fit in ws
    const size_t per = (size_t)BS * NGT * NQ * sizeof(float);   // 128 KB per slice
    int nsl = 16;
    while (nsl > 1 && (size_t)nsl * per > ws_size) nsl >>= 1;
    const int slice_len = DIM / nsl;

    cost_kernel<<<dim3(BS, nsl), 256, 0, stream>>>(heat, gt, part, slice_len);
    lsap_kernel<<<BS, 32, 0, stream>>>(part, nsl, prob, label, out);
}